// GCN_21595095564583
// MI455X (gfx1250) — compile-verified
//
#include <hip/hip_runtime.h>
#include <hip/hip_bf16.h>
#include <stdint.h>

// ---------------------------------------------------------------------------
// GCN forward (4-layer, PyG GCNConv semantics) for MI455X / gfx1250.
//
// Dense transforms: V_WMMA_F32_16X16X4_F32, one 16-node tile per wave32,
// 4 chained K=4 WMMAs per 16x16 @ 16x16, B fragments hoisted, A fragments
// loaded as clause'd b64 vectors, D stored via 8 immediate-offset b32 stores
// (coalesced per half-wave) behind a wave-uniform tail guard.
// Edge aggregation (the roofline-bound part): global_atomic_add_f32 with
// float4 gathers + global_prefetch_b8 look-ahead.
// ---------------------------------------------------------------------------

typedef float v2f __attribute__((ext_vector_type(2)));
typedef float v8f __attribute__((ext_vector_type(8)));

__device__ __forceinline__ int imin(int a, int b) { return a < b ? a : b; }

// ---------------- degree / normalization ----------------

__global__ void k_init_deg(float* __restrict__ deg, int n) {
    int i = blockIdx.x * blockDim.x + threadIdx.x;
    if (i < n) deg[i] = 1.0f;   // self loop
}

__global__ void k_accum_deg(const int* __restrict__ dst, float* __restrict__ deg, int e) {
    int i = blockIdx.x * blockDim.x + threadIdx.x;
    if (i < e) atomicAdd(&deg[dst[i]], 1.0f);
}

__global__ void k_rsqrt(float* __restrict__ dinv, int n) {
    int i = blockIdx.x * blockDim.x + threadIdx.x;
    if (i < n) {
        float d = dinv[i];
        dinv[i] = d > 0.0f ? __frsqrt_rn(d) : 0.0f;
    }
}

// ---------------- WMMA dense transforms ----------------
//
// A (16x4 f32): lane L holds row M = L&15; lanes 0-15 hold K={4k,4k+1},
// lanes 16-31 hold K={4k+2,4k+3} in VGPRs {0,1}.  (ISA 7.12.2)
// B (4x16 f32): VGPR0 = rows {4k, 4k+2}, VGPR1 = rows {4k+1, 4k+3},
// cols striped over lane&15.
// C/D (16x16 f32): VGPR j holds M = j + (lane>=16 ? 8 : 0), N = lane&15.

__global__ void k_transform16_wmma(const float* __restrict__ hin,
                                   const float* __restrict__ W,   // 16x16 row-major
                                   float* __restrict__ hout,
                                   int ntiles, int nrows) {
    const int  lane = threadIdx.x & 31;
    const int  col  = lane & 15;
    const bool hi   = lane >= 16;
    const int  koff = hi ? 2 : 0;

    // B fragments: tile-invariant, hoisted (8 VGPRs).
    v2f bf[4];
#pragma unroll
    for (int k = 0; k < 4; ++k) {
        const int k0 = 4 * k + koff;
        bf[k].x = W[(size_t)(k0    ) * 16 + col];
        bf[k].y = W[(size_t)(k0 + 1) * 16 + col];
    }

    const int stride = gridDim.x * (blockDim.x >> 5);
    for (int tile = blockIdx.x * (blockDim.x >> 5) + (threadIdx.x >> 5);
         tile < ntiles; tile += stride) {
        const int base = tile * 16;
        const int arow = imin(base + col, nrows - 1);        // branchless clamp
        const float* rp = hin + (size_t)arow * 16 + koff;

        v8f c = {};
#pragma unroll
        for (int k = 0; k < 4; ++k) {
            v2f a = *(const v2f*)(rp + 4 * k);               // 8B-aligned b64 load
            c = __builtin_amdgcn_wmma_f32_16x16x4_f32(false, a, false, bf[k],
                                                      (short)0, c, false, false);
        }

        float* op = hout + ((size_t)base + (hi ? 8 : 0)) * 16 + col;
        if (base + 16 <= nrows) {                            // wave-uniform fast path
#pragma unroll
            for (int j = 0; j < 8; ++j) op[(size_t)j * 16] = c[j];
        } else {                                             // cold tail tile
#pragma unroll
            for (int j = 0; j < 8; ++j) {
                const int r = base + j + (hi ? 8 : 0);
                if (r < nrows) hout[(size_t)r * 16 + col] = c[j];
            }
        }
    }
}

// Layer 1: x is [N,3]; pad K 3->4 with a zero B row, single WMMA per tile.
// K=3 multiplies a zero B row, so its A value is don't-care (address clamped).
__global__ void k_transform_l1_wmma(const float* __restrict__ x,
                                    const float* __restrict__ W1,  // 3x16 row-major
                                    float* __restrict__ hout,
                                    int ntiles, int nrows) {
    const int  lane = threadIdx.x & 31;
    const int  col  = lane & 15;
    const bool hi   = lane >= 16;
    const int  k0   = hi ? 2 : 0;

    v2f b;
    b.x = W1[(size_t)k0 * 16 + col];                       // k0 in {0,2} < 3
    b.y = (k0 + 1 < 3) ? W1[(size_t)(k0 + 1) * 16 + col] : 0.0f;

    const size_t last = (size_t)nrows * 3 - 1;
    const int stride = gridDim.x * (blockDim.x >> 5);
    for (int tile = blockIdx.x * (blockDim.x >> 5) + (threadIdx.x >> 5);
         tile < ntiles; tile += stride) {
        const int base = tile * 16;
        const int arow = imin(base + col, nrows - 1);
        const size_t ix = (size_t)arow * 3 + k0;           // <= 3N-1, in range
        size_t iy = ix + 1; if (iy > last) iy = last;      // clamp; value unused when hi

        v2f a;
        a.x = x[ix];
        a.y = hi ? 0.0f : x[iy];

        v8f c = {};
        c = __builtin_amdgcn_wmma_f32_16x16x4_f32(false, a, false, b,
                                                  (short)0, c, false, false);

        float* op = hout + ((size_t)base + (hi ? 8 : 0)) * 16 + col;
        if (base + 16 <= nrows) {                            // wave-uniform fast path
#pragma unroll
            for (int j = 0; j < 8; ++j) op[(size_t)j * 16] = c[j];
        } else {                                             // cold tail tile
#pragma unroll
            for (int j = 0; j < 8; ++j) {
                const int r = base + j + (hi ? 8 : 0);
                if (r < nrows) hout[(size_t)r * 16 + col] = c[j];
            }
        }
    }
}

// ---------------- aggregation ----------------

// agg[i] = h_t[i] * dinv[i]^2   (self-loop contribution; also zero-inits agg)
__global__ void k_self_init(const float* __restrict__ ht, const float* __restrict__ dinv,
                            float* __restrict__ agg, int n) {
    int t = blockIdx.x * blockDim.x + threadIdx.x;   // one thread per (node, 4 feats)
    int i = t >> 2;
    if (i >= n) return;
    int f4 = (t & 3) * 4;
    float s = dinv[i]; s *= s;
    float4 v = *(const float4*)(ht + (size_t)i * 16 + f4);
    float4 o = make_float4(v.x * s, v.y * s, v.z * s, v.w * s);
    *(float4*)(agg + (size_t)i * 16 + f4) = o;
}

// agg[dst] += h_t[src] * dinv[src]*dinv[dst], 4 lanes per edge, float4 gather.
__global__ void k_edge_scatter16(const int* __restrict__ src, const int* __restrict__ dst,
                                 const float* __restrict__ ht, const float* __restrict__ dinv,
                                 float* __restrict__ agg, int e) {
    int t  = blockIdx.x * blockDim.x + threadIdx.x;
    int ei = t >> 2;
    if (ei >= e) return;
    int f4 = (t & 3) * 4;
    int s  = src[ei];
    int d  = dst[ei];

    // pull an upcoming gather row toward the caches (global_prefetch_b8)
    int epf = ei + 4096;
    if ((t & 3) == 0 && epf < e)
        __builtin_prefetch(ht + (size_t)src[epf] * 16, 0, 1);

    float nrm = dinv[s] * dinv[d];
    float4 v  = *(const float4*)(ht + (size_t)s * 16 + f4);
    float* o  = agg + (size_t)d * 16 + f4;
    atomicAdd(o + 0, v.x * nrm);
    atomicAdd(o + 1, v.y * nrm);
    atomicAdd(o + 2, v.z * nrm);
    atomicAdd(o + 3, v.w * nrm);
}

__global__ void k_bias_relu(float* __restrict__ h, const float* __restrict__ b, int n) {
    int t = blockIdx.x * blockDim.x + threadIdx.x;
    int i = t >> 2;
    if (i >= n) return;
    int f4 = (t & 3) * 4;
    float4 bb = *(const float4*)(b + f4);
    float4 v  = *(float4*)(h + (size_t)i * 16 + f4);
    v.x = fmaxf(v.x + bb.x, 0.0f);
    v.y = fmaxf(v.y + bb.y, 0.0f);
    v.z = fmaxf(v.z + bb.z, 0.0f);
    v.w = fmaxf(v.w + bb.w, 0.0f);
    *(float4*)(h + (size_t)i * 16 + f4) = v;
}

// ---------------- layer 4 (16 -> 1) + log_softmax ----------------

__global__ void k_final_transform(const float* __restrict__ h, const float* __restrict__ W4,
                                  float* __restrict__ t4, int n) {
    int i = blockIdx.x * blockDim.x + threadIdx.x;
    if (i >= n) return;
    const float4* r = (const float4*)(h + (size_t)i * 16);
    const float4* w = (const float4*)W4;
    float acc = 0.0f;
#pragma unroll
    for (int q = 0; q < 4; ++q) {
        float4 a = r[q], b = w[q];
        acc += a.x * b.x + a.y * b.y + a.z * b.z + a.w * b.w;
    }
    t4[i] = acc;
}

__global__ void k_final_init(const float* __restrict__ t4, const float* __restrict__ dinv,
                             float* __restrict__ agg4, int n) {
    int i = blockIdx.x * blockDim.x + threadIdx.x;
    if (i >= n) return;
    float s = dinv[i];
    agg4[i] = t4[i] * s * s;
}

__global__ void k_final_scatter(const int* __restrict__ src, const int* __restrict__ dst,
                                const float* __restrict__ t4, const float* __restrict__ dinv,
                                float* __restrict__ agg4, int e) {
    int i = blockIdx.x * blockDim.x + threadIdx.x;
    if (i >= e) return;
    int s = src[i], d = dst[i];
    atomicAdd(&agg4[d], t4[s] * dinv[s] * dinv[d]);
}

// log_softmax over a width-1 axis: v - logsumexp(v) == v - v (exact, NaN/Inf-safe)
__global__ void k_final_out(const float* __restrict__ agg4, const float* __restrict__ b4,
                            float* __restrict__ out, int n) {
    int i = blockIdx.x * blockDim.x + threadIdx.x;
    if (i >= n) return;
    float v = agg4[i] + b4[0];
    out[i] = v - v;
}

// ---------------- orchestration ----------------

static inline int cdiv(long long a, long long b) { return (int)((a + b - 1) / b); }

extern "C" void kernel_launch(void* const* d_in, const int* in_sizes, int n_in,
                              void* d_out, int out_size, void* d_ws, size_t ws_size,
                              hipStream_t stream) {
    const int N = in_sizes[0] / 3;
    const int E = in_sizes[1] / 2;

    const float* x   = (const float*)d_in[0];
    const int*   ei  = (const int*)d_in[1];      // [2,E] flat, int32 (JAX x64 off)
    const int*   src = ei;
    const int*   dst = ei + E;
    const float* W1 = (const float*)d_in[2];
    const float* b1 = (const float*)d_in[3];
    const float* W2 = (const float*)d_in[4];
    const float* b2 = (const float*)d_in[5];
    const float* W3 = (const float*)d_in[6];
    const float* b3 = (const float*)d_in[7];
    const float* W4 = (const float*)d_in[8];
    const float* b4 = (const float*)d_in[9];

    // workspace layout (all 16B aligned for N multiple of 4)
    char*  ws   = (char*)d_ws;
    float* dinv = (float*)ws;                              // N
    float* hT   = (float*)(ws + (size_t)N * 4);            // N*16 (transform out)
    float* hC   = hT + (size_t)N * 16;                     // N*16 (layer out / agg)
    float* t4   = hC + (size_t)N * 16;                     // N
    float* agg4 = t4 + N;                                  // N

    const int B      = 256;
    const int ntiles = cdiv(N, 16);
    const int gTile  = cdiv(ntiles, B / 32);   // one wave per 16-node tile

    // degrees -> dinv
    k_init_deg <<<cdiv(N, B), B, 0, stream>>>(dinv, N);
    k_accum_deg<<<cdiv(E, B), B, 0, stream>>>(dst, dinv, E);
    k_rsqrt    <<<cdiv(N, B), B, 0, stream>>>(dinv, N);

    const float* Ws[3] = {W1, W2, W3};
    const float* bs[3] = {b1, b2, b3};
    for (int l = 0; l < 3; ++l) {
        if (l == 0)
            k_transform_l1_wmma<<<gTile, B, 0, stream>>>(x, W1, hT, ntiles, N);
        else
            k_transform16_wmma <<<gTile, B, 0, stream>>>(hC, Ws[l], hT, ntiles, N);
        k_self_init     <<<cdiv((long long)N * 4, B), B, 0, stream>>>(hT, dinv, hC, N);
        k_edge_scatter16<<<cdiv((long long)E * 4, B), B, 0, stream>>>(src, dst, hT, dinv, hC, E);
        k_bias_relu     <<<cdiv((long long)N * 4, B), B, 0, stream>>>(hC, bs[l], N);
    }

    k_final_transform<<<cdiv(N, B), B, 0, stream>>>(hC, W4, t4, N);
    k_final_init     <<<cdiv(N, B), B, 0, stream>>>(t4, dinv, agg4, N);
    k_final_scatter  <<<cdiv(E, B), B, 0, stream>>>(src, dst, t4, dinv, agg4, E);
    k_final_out      <<<cdiv(N, B), B, 0, stream>>>(agg4, b4, (float*)d_out, N);
}